// Feature_1657857376256
// MI455X (gfx1250) — compile-verified
//
#include <hip/hip_runtime.h>
#include <math.h>

#define B_      8
#define C_      128
#define NSOLU_  32
#define NR_     16
#define NA_     8
#define RCR_    5.2f
#define RCA_    3.5f
#define PI_     3.14159265358979323846f
#define LOG2E_  1.44269504088896341f

typedef __attribute__((ext_vector_type(2))) float v2f;
typedef __attribute__((ext_vector_type(8))) float v8f;

// ---------------------------------------------------------------------------
// Kernel 1: per-batch Gram matrix G = Xc * Xc^T via V_WMMA_F32_16X16X4_F32.
// Xc = coords - box/2 (centering: every quantity we later derive from G is
// translation invariant, so centering only improves f32 conditioning).
// One wave computes one 16x16 tile of G; grid = (64 tiles, 8 batches).
// ---------------------------------------------------------------------------
__global__ void __launch_bounds__(32) gram_wmma(const float* __restrict__ coords,
                                                float* __restrict__ G) {
  const int b    = blockIdx.y;
  const int tm   = blockIdx.x >> 3;   // row tile
  const int tn   = blockIdx.x & 7;    // col tile
  const int lane = threadIdx.x;       // wave32
  const int half = lane >> 4;
  const int l    = lane & 15;

  const float* cb = coords + b * C_ * 3;
  const int am = tm * 16 + l;         // A-matrix row (M)
  const int bn = tn * 16 + l;         // B-matrix col (N)

  // 32-bit A 16x4 layout: lanes 0-15 -> (K=0,K=1), lanes 16-31 -> (K=2,K=3).
  // B 4x16 mirrors it with N striped across the lane halves.
  v2f A, Bv;
  if (half == 0) {
    A.x  = cb[am * 3 + 0] - 6.0f;  A.y  = cb[am * 3 + 1] - 6.0f;
    Bv.x = cb[bn * 3 + 0] - 6.0f;  Bv.y = cb[bn * 3 + 1] - 6.0f;
  } else {
    A.x  = cb[am * 3 + 2] - 6.0f;  A.y  = 0.0f;   // K=3 pad
    Bv.x = cb[bn * 3 + 2] - 6.0f;  Bv.y = 0.0f;
  }

  v8f acc = {};
  acc = __builtin_amdgcn_wmma_f32_16x16x4_f32(
      /*neg_a=*/false, A, /*neg_b=*/false, Bv,
      /*c_mod=*/(short)0, acc, /*reuse_a=*/false, /*reuse_b=*/false);

  // C/D layout: VGPR v: lanes 0-15 -> M=v, lanes 16-31 -> M=v+8; N = lane&15.
  float* Gb = G + b * C_ * C_;
  const int row0 = tm * 16 + half * 8;
  const int col  = tn * 16 + l;
#pragma unroll
  for (int v = 0; v < 8; ++v) {
    Gb[(row0 + v) * C_ + col] = acc[v];
  }
}

// ---------------------------------------------------------------------------
// Kernel 2: one block (256 threads = 8 wave32) per (batch, solute atom).
// All geometry is derived from the Gram matrix:
//   d_ij^2      = G_ii - 2 G_ij + G_jj
//   d_j1j2^2    = G_j1j1 - 2 G_j1j2 + G_j2j2
//   r12 . r13   = G_ii - G_ij1 - G_ij2 + G_j1j2
// ---------------------------------------------------------------------------
__global__ void __launch_bounds__(256) acsf_kernel(
    const float* __restrict__ EtaR_tab, const float* __restrict__ ShfR_tab,
    const float* __restrict__ Zeta_tab, const float* __restrict__ EtaA_tab,
    const int*   __restrict__ atom_types,
    const float* __restrict__ G,
    float*       __restrict__ out) {
  const int blk = blockIdx.x;
  const int b   = blk >> 5;      // / NSOLU_
  const int i   = blk & 31;
  const int tid = threadIdx.x;

  __shared__ float sGi[C_];      // G[i, j(k)]
  __shared__ float sGd[C_];      // G[j(k), j(k)]
  __shared__ float sD2[C_];      // d_ij^2
  __shared__ float sFA[C_];      // fcut(d_ij, RCA)
  __shared__ float sG2[NR_];
  __shared__ float sG3[NA_];
  __shared__ float sEtaR[NR_], sShfR[NR_], sZeta[NA_], sEtaA[NA_];
  __shared__ float sInv2, sInv3;

  const int t = atom_types[b * NSOLU_ + i];
  if (tid < NR_) {
    sEtaR[tid] = EtaR_tab[t * NR_ + tid];
    sShfR[tid] = ShfR_tab[t * NR_ + tid];
    sG2[tid]   = 0.0f;
  }
  if (tid < NA_) {
    sZeta[tid] = Zeta_tab[t * NA_ + tid];
    sEtaA[tid] = EtaA_tab[t * NA_ + tid];
    sG3[tid]   = 0.0f;
  }
  const float* Gb  = G + b * C_ * C_;
  const float  Gii = Gb[i * C_ + i];
  __syncthreads();

  const int M = C_ - 1;          // 127 neighbors (j != i)

  // ---- Phase 1: per-neighbor prep + radial (G2) features ------------------
  if (tid < M) {
    const int   j   = (tid < i) ? tid : tid + 1;
    const float Gij = Gb[i * C_ + j];
    const float Gjj = Gb[j * C_ + j];
    const float d2  = fmaxf(Gii - 2.0f * Gij + Gjj, 0.0f);
    const float d   = sqrtf(d2);
    sGi[tid] = Gij;
    sGd[tid] = Gjj;
    sD2[tid] = d2;
    sFA[tid] = (d < RCA_) ? 0.5f * (__cosf(d * (PI_ / RCA_)) + 1.0f) : 0.0f;
    const float fR = (d < RCR_) ? 0.5f * (__cosf(d * (PI_ / RCR_)) + 1.0f) : 0.0f;
    if (fR > 0.0f) {
#pragma unroll
      for (int r = 0; r < NR_; ++r) {
        const float dd = d - sShfR[r];
        atomicAdd(&sG2[r], exp2f(-sEtaR[r] * dd * dd * LOG2E_) * fR);
      }
    }
  }
  __syncthreads();

  // ---- Phase 2: angular (G3) features over 8001 neighbor pairs ------------
  const int NPAIR = M * (M - 1) / 2;   // 8001
  float acc[NA_];
#pragma unroll
  for (int a = 0; a < NA_; ++a) acc[a] = 0.0f;

  for (int p = tid; p < NPAIR; p += 256) {
    // Decode p -> (k1 < k2): offset(k1) = k1*(2M-k1-1)/2
    int k1 = (int)((2.0f * M - 1.0f -
                    sqrtf((float)((2 * M - 1) * (2 * M - 1)) - 8.0f * (float)p)) * 0.5f);
    if (k1 < 0) k1 = 0;
    while ((k1 + 1) * (2 * M - k1 - 2) / 2 <= p) ++k1;
    while (k1 * (2 * M - k1 - 1) / 2 > p) --k1;
    const int k2 = k1 + 1 + (p - k1 * (2 * M - k1 - 1) / 2);

    const float f1 = sFA[k1];
    const float f2 = sFA[k2];
    if (f1 * f2 == 0.0f) continue;                 // ~99% of pairs exit here

    const int   j1   = (k1 < i) ? k1 : k1 + 1;
    const int   j2   = (k2 < i) ? k2 : k2 + 1;
    const float G12  = Gb[j1 * C_ + j2];
    const float d23q = fmaxf(sGd[k1] + sGd[k2] - 2.0f * G12, 0.0f);
    if (d23q >= RCA_ * RCA_) continue;

    const float d23   = sqrtf(d23q);
    const float f3    = 0.5f * (__cosf(d23 * (PI_ / RCA_)) + 1.0f);
    const float fprod = f1 * f2 * f3;
    const float d12q  = sD2[k1];
    const float d13q  = sD2[k2];
    const float dotv  = Gii - sGi[k1] - sGi[k2] + G12;
    const float cosv  = dotv * rsqrtf(fmaxf(d12q * d13q, 1e-30f));
    const float onec  = fmaxf(1.0f + cosv, 0.0f);
    const float L     = __log2f(onec);             // -inf at onec==0 -> term 0
    const float Sl    = (d12q + d13q + d23q) * LOG2E_;
    // 2^(1-z) * onec^z * exp(-eta*S) = exp2(1 + z*(L-1) - eta*S*log2e)
#pragma unroll
    for (int a = 0; a < NA_; ++a) {
      const float e = 1.0f + sZeta[a] * (L - 1.0f) - sEtaA[a] * Sl;
      acc[a] += exp2f(e) * fprod;
    }
  }
#pragma unroll
  for (int a = 0; a < NA_; ++a) atomicAdd(&sG3[a], acc[a]);
  __syncthreads();

  // ---- Phase 3: normalize + write -----------------------------------------
  if (tid == 0) {
    float s2 = 0.0f;
#pragma unroll
    for (int r = 0; r < NR_; ++r) s2 += sG2[r] * sG2[r];
    sInv2 = 1.0f / sqrtf(s2);
    float s3 = 0.0f;
#pragma unroll
    for (int a = 0; a < NA_; ++a) s3 += sG3[a] * sG3[a];
    sInv3 = (s3 == 0.0f) ? 1.0f : (1.0f / sqrtf(s3));
  }
  __syncthreads();

  float* o = out + (b * NSOLU_ + i) * (NR_ + NA_);
  if (tid < NR_) {
    o[tid] = sG2[tid] * sInv2;
  } else if (tid < NR_ + NA_) {
    o[tid] = sG3[tid - NR_] * sInv3;
  }
}

// ---------------------------------------------------------------------------
extern "C" void kernel_launch(void* const* d_in, const int* in_sizes, int n_in,
                              void* d_out, int out_size, void* d_ws, size_t ws_size,
                              hipStream_t stream) {
  const float* coords = (const float*)d_in[0];
  const float* EtaR   = (const float*)d_in[1];
  const float* ShfR   = (const float*)d_in[2];
  const float* Zeta   = (const float*)d_in[3];
  const float* EtaA   = (const float*)d_in[4];
  const int*   types  = (const int*)d_in[5];
  // d_in[6] = nsolu scalar (device); shapes are compile-time constants here.

  float* G   = (float*)d_ws;          // 8 * 128 * 128 * 4B = 512 KB scratch
  float* out = (float*)d_out;

  dim3 g1(64, B_);                    // 64 16x16 tiles per batch, 8 batches
  gram_wmma<<<g1, 32, 0, stream>>>(coords, G);

  acsf_kernel<<<B_ * NSOLU_, 256, 0, stream>>>(EtaR, ShfR, Zeta, EtaA, types, G, out);
}